// Head_11252814315968
// MI455X (gfx1250) — compile-verified
//
#include <hip/hip_runtime.h>
#include <hip/hip_bf16.h>

typedef __attribute__((ext_vector_type(2))) float v2f;
typedef __attribute__((ext_vector_type(8))) float v8f;

#define CCH   64     // channels C
#define GGR   128    // graphs G
#define HHID  128    // hidden H
#define LCONV 8      // layers L
#define BN_EPS 1e-5f

// ---------------------------------------------------------------- utilities
__global__ void zero_kernel(float* __restrict__ p, int n) {
  int i = blockIdx.x * blockDim.x + threadIdx.x;
  if (i < n) p[i] = 0.0f;
}

__global__ void init_deg_kernel(float* __restrict__ deg, int n) {
  int i = blockIdx.x * blockDim.x + threadIdx.x;
  if (i < n) deg[i] = 1.0f;   // self-loop contribution
}

__global__ void edge_deg_kernel(const int* __restrict__ col,
                                float* __restrict__ deg, int e) {
  int i = blockIdx.x * blockDim.x + threadIdx.x;
  if (i < e) atomicAdd(&deg[col[i]], 1.0f);
}

__global__ void deg_to_dis_kernel(float* __restrict__ deg, int n) {
  int i = blockIdx.x * blockDim.x + threadIdx.x;
  if (i < n) deg[i] = rsqrtf(deg[i]);   // in-place deg -> deg^{-1/2}
}

__global__ void norm_kernel(const int* __restrict__ row,
                            const int* __restrict__ col,
                            const float* __restrict__ dis,
                            float* __restrict__ nrm, int e) {
  int i = blockIdx.x * blockDim.x + threadIdx.x;
  if (i < e) nrm[i] = dis[row[i]] * dis[col[i]];
}

// ------------------------------------------- embedding gather + BN stats (layer 0)
__global__ __launch_bounds__(256)
void embed_stats_kernel(const int* __restrict__ x,
                        const float* __restrict__ emb,
                        float* __restrict__ h,
                        float* __restrict__ stats, int n) {
  int tid  = blockIdx.x * blockDim.x + threadIdx.x;  // 65536 threads
  int ch   = tid & (CCH - 1);
  int node = tid >> 6;                               // 0..1023
  float s = 0.0f, s2 = 0.0f;
  for (int i = node; i < n; i += 1024) {
    float v = emb[x[i] * CCH + ch];
    h[i * CCH + ch] = v;
    s += v; s2 += v * v;
  }
  atomicAdd(&stats[ch], s);
  atomicAdd(&stats[CCH + ch], s2);
}

// -------- fold BN into the GEMM: W' = diag(rsqrt(var+eps)*gamma) @ W,
// -------- d = (beta - mean*a) @ W + conv_b.  One block. Also re-zeroes stats.
__global__ __launch_bounds__(128)
void prep_kernel(float* __restrict__ stats,
                 const float* __restrict__ gamma,
                 const float* __restrict__ beta,
                 const float* __restrict__ W,
                 const float* __restrict__ cb,
                 float* __restrict__ Wp,
                 float* __restrict__ dvec, int n) {
  __shared__ float a_s[CCH], b_s[CCH];
  int t = threadIdx.x;
  if (t < CCH) {
    float inv_n = 1.0f / (float)n;
    float mean = stats[t] * inv_n;
    float var  = stats[CCH + t] * inv_n - mean * mean;
    float a    = rsqrtf(var + BN_EPS) * gamma[t];
    a_s[t] = a;
    b_s[t] = beta[t] - mean * a;
  }
  __syncthreads();
  if (t < 2 * CCH) stats[t] = 0.0f;      // ready for next layer's accumulation
  for (int idx = t; idx < CCH * CCH; idx += 128) {
    int k = idx >> 6;
    Wp[idx] = a_s[k] * W[idx];
  }
  if (t < CCH) {
    float d = cb[t];
    #pragma unroll
    for (int k = 0; k < CCH; ++k) d += b_s[k] * W[k * CCH + t];
    dvec[t] = d;
  }
}

// ------------------------------------------- hw = h @ W' + d  via fp32 WMMA
// 4 waves / block; wave w computes the 16x16 tile (rows 16*bid, cols 16*w).
__global__ __launch_bounds__(128)
void gemm_wmma_kernel(const float* __restrict__ h,
                      const float* __restrict__ Wp,
                      const float* __restrict__ dvec,
                      float* __restrict__ hw, int n) {
  int wave  = threadIdx.x >> 5;
  int lane  = threadIdx.x & 31;
  int m0    = blockIdx.x * 16;
  int n0    = wave * 16;
  int l15   = lane & 15;        // M index for A, N index for B/D
  int khalf = lane >> 4;        // selects K pair {0,1} vs {2,3} within a 4-slice

  v8f c;
  float dj = dvec[n0 + l15];
  #pragma unroll
  for (int i = 0; i < 8; ++i) c[i] = dj;   // bias-seeded accumulator

  int mA = m0 + l15; if (mA > n - 1) mA = n - 1;   // clamp: EXEC stays all-ones
  const float* arow = h + (size_t)mA * CCH;

  #pragma unroll
  for (int k = 0; k < CCH; k += 4) {
    v2f a, b;
    a[0] = arow[k + 2 * khalf];
    a[1] = arow[k + 2 * khalf + 1];
    b[0] = Wp[(k + 2 * khalf) * CCH + n0 + l15];
    b[1] = Wp[(k + 2 * khalf + 1) * CCH + n0 + l15];
    c = __builtin_amdgcn_wmma_f32_16x16x4_f32(false, a, false, b,
                                              (short)0, c, false, false);
  }

  #pragma unroll
  for (int v = 0; v < 8; ++v) {
    int m = m0 + v + 8 * khalf;
    if (m < n) hw[(size_t)m * CCH + n0 + l15] = c[v];
  }
}

// ------------------------------------------- agg init with self-loop term
__global__ __launch_bounds__(256)
void selfloop_kernel(const float* __restrict__ hw,
                     const float* __restrict__ dis,
                     float* __restrict__ agg, int n) {
  int gid  = blockIdx.x * blockDim.x + threadIdx.x;
  int i    = gid >> 5;
  int lane = gid & 31;
  if (i >= n) return;
  float w = dis[i]; w *= w;                       // 1/deg
  const float2 v = *(const float2*)(hw + (size_t)i * CCH + lane * 2);
  float2* o = (float2*)(agg + (size_t)i * CCH + lane * 2);
  *o = make_float2(v.x * w, v.y * w);
}

// ------------------------------------------- edge gather-scale-scatter (L2 atomics)
__global__ __launch_bounds__(256)
void scatter_kernel(const int* __restrict__ row,
                    const int* __restrict__ col,
                    const float* __restrict__ nrm,
                    const float* __restrict__ hw,
                    float* __restrict__ agg, int e) {
  int gid  = blockIdx.x * blockDim.x + threadIdx.x;
  int ed   = gid >> 5;
  int lane = gid & 31;
  if (ed >= e) return;
  int r = row[ed], cN = col[ed];
  float w = nrm[ed];
  const float2 v = *(const float2*)(hw + (size_t)r * CCH + lane * 2);
  atomicAdd(&agg[(size_t)cN * CCH + lane * 2],     v.x * w);
  atomicAdd(&agg[(size_t)cN * CCH + lane * 2 + 1], v.y * w);
}

// ------------------------------------------- h = relu(h+agg), fused BN stats
__global__ __launch_bounds__(256)
void update_stats_kernel(float* __restrict__ h,
                         const float* __restrict__ agg,
                         float* __restrict__ stats, int n) {
  int tid  = blockIdx.x * blockDim.x + threadIdx.x;  // 65536 threads
  int ch   = tid & (CCH - 1);
  int node = tid >> 6;
  float s = 0.0f, s2 = 0.0f;
  for (int i = node; i < n; i += 1024) {
    size_t idx = (size_t)i * CCH + ch;
    float v = h[idx] + agg[idx];
    v = v > 0.0f ? v : 0.0f;
    h[idx] = v;
    s += v; s2 += v * v;
  }
  atomicAdd(&stats[ch], s);
  atomicAdd(&stats[CCH + ch], s2);
}

// ------------------------------------------- global mean pool (scatter phase)
__global__ __launch_bounds__(256)
void pool_kernel(const float* __restrict__ h,
                 const int* __restrict__ batch,
                 float* __restrict__ pooled,
                 float* __restrict__ counts, int n) {
  int gid  = blockIdx.x * blockDim.x + threadIdx.x;
  int i    = gid >> 5;
  int lane = gid & 31;
  if (i >= n) return;
  int g = batch[i];
  const float2 v = *(const float2*)(h + (size_t)i * CCH + lane * 2);
  atomicAdd(&pooled[(size_t)g * CCH + lane * 2],     v.x);
  atomicAdd(&pooled[(size_t)g * CCH + lane * 2 + 1], v.y);
  if (lane == 0) atomicAdd(&counts[g], 1.0f);
}

// ------------------------------------------- head MLP: one block per graph
__global__ __launch_bounds__(128)
void mlp_kernel(const float* __restrict__ pooled,
                const float* __restrict__ counts,
                const float* __restrict__ hid_w,
                const float* __restrict__ hid_b,
                const float* __restrict__ out_w,
                const float* __restrict__ out_b,
                float* __restrict__ out) {
  __shared__ float p_s[CCH];
  __shared__ float hid_s[HHID];
  int g = blockIdx.x;
  int t = threadIdx.x;
  float inv_cnt = 1.0f / fmaxf(counts[g], 1.0f);
  if (t < CCH) p_s[t] = pooled[(size_t)g * CCH + t] * inv_cnt;
  __syncthreads();
  float acc = hid_b[t];
  #pragma unroll 8
  for (int k = 0; k < CCH; ++k) acc += p_s[k] * hid_w[k * HHID + t];
  hid_s[t] = fmaxf(acc, 0.0f);
  __syncthreads();
  if (t < CCH) {
    float o = out_b[t];
    #pragma unroll 8
    for (int k = 0; k < HHID; ++k) o += hid_s[k] * out_w[k * CCH + t];
    out[(size_t)g * CCH + t] = o;
  }
}

// ---------------------------------------------------------------- launcher
extern "C" void kernel_launch(void* const* d_in, const int* in_sizes, int n_in,
                              void* d_out, int out_size, void* d_ws, size_t ws_size,
                              hipStream_t stream) {
  const int*   x        = (const int*)  d_in[0];
  const int*   ei       = (const int*)  d_in[1];
  const int*   batch    = (const int*)  d_in[2];
  const float* emb      = (const float*)d_in[3];
  const float* bn_gamma = (const float*)d_in[4];
  const float* bn_beta  = (const float*)d_in[5];
  const float* conv_w   = (const float*)d_in[6];
  const float* conv_b   = (const float*)d_in[7];
  const float* hid_w    = (const float*)d_in[8];
  const float* hid_b    = (const float*)d_in[9];
  const float* out_w    = (const float*)d_in[10];
  const float* out_b    = (const float*)d_in[11];
  float* out = (float*)d_out;

  const int N = in_sizes[0];
  const int E = in_sizes[1] / 2;
  const int* row = ei;        // edge_index[0]
  const int* col = ei + E;    // edge_index[1]

  // workspace layout (floats)
  float* ws = (float*)d_ws;
  size_t off = 0;
  float* h      = ws + off; off += (size_t)N * CCH;
  float* hw     = ws + off; off += (size_t)N * CCH;
  float* agg    = ws + off; off += (size_t)N * CCH;
  float* dis    = ws + off; off += (size_t)N;          // deg -> deg^{-1/2}
  float* nrm    = ws + off; off += (size_t)E;
  float* stats  = ws + off; off += 2 * CCH;
  float* Wp     = ws + off; off += CCH * CCH;
  float* dvec   = ws + off; off += CCH;
  float* pooled = ws + off; off += (size_t)GGR * CCH;
  float* counts = ws + off; off += GGR;

  const int T256 = 256;
  auto blocks = [](long long work, int tpb) { return (int)((work + tpb - 1) / tpb); };

  // degree / normalization
  init_deg_kernel<<<blocks(N, T256), T256, 0, stream>>>(dis, N);
  edge_deg_kernel<<<blocks(E, T256), T256, 0, stream>>>(col, dis, E);
  deg_to_dis_kernel<<<blocks(N, T256), T256, 0, stream>>>(dis, N);
  norm_kernel<<<blocks(E, T256), T256, 0, stream>>>(row, col, dis, nrm, E);

  // embedding + layer-0 BN stats
  zero_kernel<<<1, 2 * CCH, 0, stream>>>(stats, 2 * CCH);
  embed_stats_kernel<<<256, 256, 0, stream>>>(x, emb, h, stats, N);

  // 8 GCN layers
  for (int l = 0; l < LCONV; ++l) {
    prep_kernel<<<1, 128, 0, stream>>>(stats,
                                       bn_gamma + l * CCH, bn_beta + l * CCH,
                                       conv_w + (size_t)l * CCH * CCH,
                                       conv_b + l * CCH, Wp, dvec, N);
    gemm_wmma_kernel<<<blocks(N, 16), 128, 0, stream>>>(h, Wp, dvec, hw, N);
    selfloop_kernel<<<blocks((long long)N * 32, T256), T256, 0, stream>>>(hw, dis, agg, N);
    scatter_kernel<<<blocks((long long)E * 32, T256), T256, 0, stream>>>(row, col, nrm, hw, agg, E);
    update_stats_kernel<<<256, 256, 0, stream>>>(h, agg, stats, N);
  }

  // pooling + head MLP
  zero_kernel<<<blocks(GGR * CCH + GGR, T256), T256, 0, stream>>>(pooled, GGR * CCH + GGR);
  pool_kernel<<<blocks((long long)N * 32, T256), T256, 0, stream>>>(h, batch, pooled, counts, N);
  mlp_kernel<<<GGR, HHID, 0, stream>>>(pooled, counts, hid_w, hid_b, out_w, out_b, out);
}